// DigitCaps_34248069218618
// MI455X (gfx1250) — compile-verified
//
#include <hip/hip_runtime.h>
#include <hip/hip_bf16.h>
#include <math.h>

typedef __attribute__((ext_vector_type(2))) float v2f;
typedef __attribute__((ext_vector_type(8))) float v8f;

#define B_TOT   256
#define N_CAPS  2304
#define IN_DIMS 8
#define NCLS    10
#define OUT_D   16
#define CD      160        // NCLS * OUT_D
#define RBIAS   0.1f

#define NBLK       36                               // blocks per batch-tile
#define WAVES_BLK  8
#define N_PER_WAVE (N_CAPS / (NBLK * WAVES_BLK))    // 2304 / 288 = 8

// ---------------------------------------------------------------------------
// Pass A: s[b,c,d] += softmax_c(blog[b,n,:]) * u_hat[b,n,c,d]   (u_hat via WMMA)
// ITER0: softmax(0) == 1/10 uniform -> skip blog reads / shuffles, scale once.
// ---------------------------------------------------------------------------
template <bool ITER0>
__global__ __launch_bounds__(256) void caps_passA(const float* __restrict__ x,
                                                  const float* __restrict__ W,
                                                  const float* __restrict__ blog,
                                                  float* __restrict__ s_glob)
{
    __shared__ float s_blk[16 * CD];

    const int tid = threadIdx.x;
    for (int i = tid; i < 16 * CD; i += 256) s_blk[i] = 0.0f;
    __syncthreads();

    const int btBase = blockIdx.y * 16;
    const int wave = tid >> 5;
    const int ln   = tid & 31;
    const int lo   = ln & 15;        // d index / batch row for loads
    const int hi   = ln >> 4;        // half-wave select

    const int wglob = blockIdx.x * WAVES_BLK + wave;   // 0..287 within btile

    v8f s_acc[NCLS];
    #pragma unroll
    for (int c = 0; c < NCLS; ++c)
        #pragma unroll
        for (int r = 0; r < 8; ++r) s_acc[c][r] = 0.0f;

    for (int k = 0; k < N_PER_WAVE; ++k) {
        const int n = wglob * N_PER_WAVE + k;

        // A-matrix (16x8 batch tile of x), ISA layout: v[j] = x[b=lo, i = 2*hi + j (+4)]
        const float* xb = x + (size_t)(btBase + lo) * (N_CAPS * IN_DIMS)
                            + (size_t)n * IN_DIMS + 2 * hi;
        v2f A0, A1;
        A0.x = xb[0]; A0.y = xb[1];
        A1.x = xb[4]; A1.y = xb[5];

        // softmax coefficients for this n (lane lo owns batch row btBase+lo)
        float coef[NCLS];
        if (!ITER0) {
            const float* bl = blog + (size_t)(btBase + lo) * (N_CAPS * NCLS)
                                   + (size_t)n * NCLS;
            float v[NCLS];
            float m = bl[0];
            #pragma unroll
            for (int c = 0; c < NCLS; ++c) { v[c] = bl[c]; m = fmaxf(m, v[c]); }
            float sum = 0.0f;
            #pragma unroll
            for (int c = 0; c < NCLS; ++c) { v[c] = __expf(v[c] - m); sum += v[c]; }
            const float inv = 1.0f / sum;
            #pragma unroll
            for (int c = 0; c < NCLS; ++c) coef[c] = v[c] * inv;
        }

        // B-matrix rows of W[n]: v[j] = W[n, i = 2*hi + j (+4), 16c + lo]
        const float* wn = W + (size_t)n * (IN_DIMS * CD) + (size_t)(2 * hi) * CD + lo;

        #pragma unroll
        for (int c = 0; c < NCLS; ++c) {
            v2f B0, B1;
            B0.x = wn[c * 16];                B0.y = wn[c * 16 + CD];
            B1.x = wn[c * 16 + 4 * CD];       B1.y = wn[c * 16 + 5 * CD];

            v8f acc;
            #pragma unroll
            for (int r = 0; r < 8; ++r) acc[r] = 0.0f;
            acc = __builtin_amdgcn_wmma_f32_16x16x4_f32(false, A0, false, B0,
                                                        (short)0, acc, false, false);
            acc = __builtin_amdgcn_wmma_f32_16x16x4_f32(false, A1, false, B1,
                                                        (short)0, acc, false, false);
            // acc[r] @ lane ln == u_hat[b = r + 8*hi, n, c, d = lo]
            if (ITER0) {
                #pragma unroll
                for (int r = 0; r < 8; ++r) s_acc[c][r] += acc[r];
            } else {
                #pragma unroll
                for (int r = 0; r < 8; ++r) {
                    float cf = __shfl(coef[c], r + 8 * hi, 32);
                    s_acc[c][r] += cf * acc[r];
                }
            }
        }
    }

    if (ITER0) {
        #pragma unroll
        for (int c = 0; c < NCLS; ++c)
            #pragma unroll
            for (int r = 0; r < 8; ++r) s_acc[c][r] *= 0.1f;   // softmax(0) = 1/10
    }

    // reduce 8 waves into LDS
    #pragma unroll
    for (int c = 0; c < NCLS; ++c)
        #pragma unroll
        for (int r = 0; r < 8; ++r)
            atomicAdd(&s_blk[(r + 8 * hi) * CD + c * 16 + lo], s_acc[c][r]);
    __syncthreads();

    // block partial -> global s (s pre-initialized to ROUTING_BIAS)
    for (int i = tid; i < 16 * CD; i += 256) {
        const int b = i / CD;
        atomicAdd(&s_glob[(size_t)(btBase + b) * CD + (i % CD)], s_blk[i]);
    }
}

// ---------------------------------------------------------------------------
// Pass B: blog[b,n,c] = (FIRST ? 0 : blog) + sum_d u_hat[b,n,c,d] * s[b,c,d]
// ---------------------------------------------------------------------------
template <bool FIRST>
__global__ __launch_bounds__(256) void caps_passB(const float* __restrict__ x,
                                                  const float* __restrict__ W,
                                                  const float* __restrict__ s_glob,
                                                  float* __restrict__ blog)
{
    __shared__ float s_lds[16 * CD];

    const int tid = threadIdx.x;
    const int btBase = blockIdx.y * 16;
    for (int i = tid; i < 16 * CD; i += 256)
        s_lds[i] = s_glob[(size_t)btBase * CD + i];
    __syncthreads();

    const int wave = tid >> 5;
    const int ln   = tid & 31;
    const int lo   = ln & 15;
    const int hi   = ln >> 4;
    const int wglob = blockIdx.x * WAVES_BLK + wave;

    for (int k = 0; k < N_PER_WAVE; ++k) {
        const int n = wglob * N_PER_WAVE + k;

        const float* xb = x + (size_t)(btBase + lo) * (N_CAPS * IN_DIMS)
                            + (size_t)n * IN_DIMS + 2 * hi;
        v2f A0, A1;
        A0.x = xb[0]; A0.y = xb[1];
        A1.x = xb[4]; A1.y = xb[5];

        const float* wn = W + (size_t)n * (IN_DIMS * CD) + (size_t)(2 * hi) * CD + lo;

        #pragma unroll
        for (int c = 0; c < NCLS; ++c) {
            v2f B0, B1;
            B0.x = wn[c * 16];                B0.y = wn[c * 16 + CD];
            B1.x = wn[c * 16 + 4 * CD];       B1.y = wn[c * 16 + 5 * CD];

            v8f acc;
            #pragma unroll
            for (int r = 0; r < 8; ++r) acc[r] = 0.0f;
            acc = __builtin_amdgcn_wmma_f32_16x16x4_f32(false, A0, false, B0,
                                                        (short)0, acc, false, false);
            acc = __builtin_amdgcn_wmma_f32_16x16x4_f32(false, A1, false, B1,
                                                        (short)0, acc, false, false);

            #pragma unroll
            for (int r = 0; r < 8; ++r) {
                // product for d = lo, then butterfly-sum over the 16 d lanes
                float p = acc[r] * s_lds[(r + 8 * hi) * CD + c * 16 + lo];
                p += __shfl_xor(p, 1, 32);
                p += __shfl_xor(p, 2, 32);
                p += __shfl_xor(p, 4, 32);
                p += __shfl_xor(p, 8, 32);
                if (lo == r) {   // lane r writes b=r (hi=0), lane 16+r writes b=8+r
                    size_t a = (size_t)(btBase + r + 8 * hi) * (N_CAPS * NCLS)
                             + (size_t)n * NCLS + c;
                    if (FIRST) blog[a] = p;
                    else       blog[a] = blog[a] + p;
                }
            }
        }
    }
}

// ---------------------------------------------------------------------------
__global__ void caps_init_s(float* __restrict__ s)
{
    const int i = blockIdx.x * 256 + threadIdx.x;
    if (i < B_TOT * CD) s[i] = RBIAS;
}

__global__ void caps_squash(const float* __restrict__ s_glob, float* __restrict__ out)
{
    const int i = blockIdx.x * 256 + threadIdx.x;   // lanes 0..15 = d of one (b,c) row
    if (i >= B_TOT * CD) return;
    float sv = s_glob[i];
    float sq = sv * sv;
    sq += __shfl_xor(sq, 1, 32);
    sq += __shfl_xor(sq, 2, 32);
    sq += __shfl_xor(sq, 4, 32);
    sq += __shfl_xor(sq, 8, 32);
    const float norm = sqrtf(sq);
    out[i] = sv * norm / (1.0f + sq);
}

// ---------------------------------------------------------------------------
extern "C" void kernel_launch(void* const* d_in, const int* in_sizes, int n_in,
                              void* d_out, int out_size, void* d_ws, size_t ws_size,
                              hipStream_t stream)
{
    (void)in_sizes; (void)n_in; (void)out_size; (void)ws_size;
    const float* x = (const float*)d_in[0];   // [256, 2304, 8]
    const float* W = (const float*)d_in[1];   // [2304, 8, 160]
    float* out = (float*)d_out;               // [256, 10, 16]

    float* s_glob = (float*)d_ws;             // 256*160 floats
    float* blog   = s_glob + B_TOT * CD;      // 256*2304*10 floats (~23.6 MB)

    const dim3 grid(NBLK, B_TOT / 16);
    const dim3 blk(256);
    const dim3 g1((B_TOT * CD + 255) / 256);

    // iter 0
    caps_init_s<<<g1, blk, 0, stream>>>(s_glob);
    caps_passA<true><<<grid, blk, 0, stream>>>(x, W, blog, s_glob);
    caps_passB<true><<<grid, blk, 0, stream>>>(x, W, s_glob, blog);
    // iter 1
    caps_init_s<<<g1, blk, 0, stream>>>(s_glob);
    caps_passA<false><<<grid, blk, 0, stream>>>(x, W, blog, s_glob);
    caps_passB<false><<<grid, blk, 0, stream>>>(x, W, s_glob, blog);
    // iter 2 (final): s then squash
    caps_init_s<<<g1, blk, 0, stream>>>(s_glob);
    caps_passA<false><<<grid, blk, 0, stream>>>(x, W, blog, s_glob);
    caps_squash<<<g1, blk, 0, stream>>>(s_glob, out);
}